// TransitionDown_16716012716553
// MI455X (gfx1250) — compile-verified
//
#include <hip/hip_runtime.h>
#include <math.h>

typedef float v2f __attribute__((ext_vector_type(2)));
typedef float v8f __attribute__((ext_vector_type(8)));

#define NBATCH 8
#define NPTS   4096
#define CIN    64
#define COUT   128
#define NSAMP  1024
#define KNNK   16
#define EPSV   1e-5f

// ---------------------------------------------------------------------------
// Kernel 0: zero the BN statistic accumulators (ws is poisoned, not zeroed)
// ---------------------------------------------------------------------------
__global__ void init_stats_kernel(float* __restrict__ gsum, float* __restrict__ gsq) {
  int t = threadIdx.x;
  if (t < COUT) { gsum[t] = 0.0f; gsq[t] = 0.0f; }
}

// ---------------------------------------------------------------------------
// Kernel 1: farthest point sampling. One 256-thread block per batch.
// Each thread owns 16 points entirely in registers; per-step argmax via
// wave32 shuffle butterfly + 8-entry LDS cross-wave reduce.
// ---------------------------------------------------------------------------
__global__ __launch_bounds__(256) void fps_kernel(const float* __restrict__ pos,
                                                  int* __restrict__ fps_idx) {
  const int b = blockIdx.x;
  const int t = threadIdx.x;
  const int lane = t & 31;
  const int wv = t >> 5;
  const float* pb = pos + (size_t)b * NPTS * 3;

  float px[16], py[16], pz[16], md[16];
#pragma unroll
  for (int j = 0; j < 16; ++j) {
    int i = j * 256 + t;
    px[j] = pb[i * 3 + 0];
    py[j] = pb[i * 3 + 1];
    pz[j] = pb[i * 3 + 2];
    md[j] = 1e10f;
  }

  __shared__ float s_lx, s_ly, s_lz;
  __shared__ float s_bd[8];
  __shared__ int   s_bi[8];
  __shared__ int   s_next;

  int last = 0;
  for (int s = 0; s < NSAMP; ++s) {
    // owner thread broadcasts coords of current `last`
    if (t == (last & 255)) {
      int j = last >> 8;
      s_lx = px[j]; s_ly = py[j]; s_lz = pz[j];
    }
    if (t == 0) fps_idx[b * NSAMP + s] = last;
    __syncthreads();
    const float lx = s_lx, ly = s_ly, lz = s_lz;

    float bestd = -1.0f; int besti = 0x7fffffff;
#pragma unroll
    for (int j = 0; j < 16; ++j) {
      float dx = px[j] - lx, dy = py[j] - ly, dz = pz[j] - lz;
      float d = dx * dx + dy * dy + dz * dz;
      float m = fminf(md[j], d);
      md[j] = m;
      if (m > bestd) { bestd = m; besti = j * 256 + t; }  // j ascending -> first max kept
    }
    // wave32 argmax reduction, ties -> smaller index (matches jnp.argmax)
#pragma unroll
    for (int off = 16; off > 0; off >>= 1) {
      float od = __shfl_xor(bestd, off, 32);
      int   oi = __shfl_xor(besti, off, 32);
      if (od > bestd || (od == bestd && oi < besti)) { bestd = od; besti = oi; }
    }
    if (lane == 0) { s_bd[wv] = bestd; s_bi[wv] = besti; }
    __syncthreads();
    if (t == 0) {
      float bd = s_bd[0]; int bi = s_bi[0];
#pragma unroll
      for (int w = 1; w < 8; ++w) {
        if (s_bd[w] > bd || (s_bd[w] == bd && s_bi[w] < bi)) { bd = s_bd[w]; bi = s_bi[w]; }
      }
      s_next = bi;
    }
    __syncthreads();
    last = s_next;
  }
}

// ---------------------------------------------------------------------------
// Kernel 2: h = x @ W^T + b via V_WMMA_F32_16X16X4_F32 (full f32 precision),
// fused per-channel sum / sum-of-squares for BatchNorm (LDS + global atomics).
// One wave per 16x16 output tile; 16 K-steps of K=4.
// ---------------------------------------------------------------------------
__global__ __launch_bounds__(256) void gemm_stats_kernel(const float* __restrict__ x,
                                                         const float* __restrict__ W,
                                                         const float* __restrict__ bias,
                                                         float* __restrict__ h,
                                                         float* __restrict__ gsum,
                                                         float* __restrict__ gsq) {
  const int lane  = threadIdx.x & 31;
  const int wv    = threadIdx.x >> 5;
  const int ntile = blockIdx.x;             // 0..7   (COUT/16)
  const int mtile = blockIdx.y * 8 + wv;    // 0..2047
  const int m0 = mtile * 16;
  const int n0 = ntile * 16;

  // A 16x4 f32 layout: lanes 0-15 -> M=lane, K=kp..kp+1 with kp = (lane>>4)*2
  const int rc = lane & 15;                 // row of A / col of B
  const int kp = (lane >> 4) * 2;
  const float* xr = x + (size_t)(m0 + rc) * CIN + kp;  // A[row][kp + ...]
  const float* wr = W + (size_t)(n0 + rc) * CIN + kp;  // B[k][col] = W[col][k]

  v8f c = {};
#pragma unroll
  for (int kb = 0; kb < CIN / 4; ++kb) {
    v2f a, bm;
    a.x  = xr[kb * 4 + 0];
    a.y  = xr[kb * 4 + 1];
    bm.x = wr[kb * 4 + 0];
    bm.y = wr[kb * 4 + 1];
    c = __builtin_amdgcn_wmma_f32_16x16x4_f32(false, a, false, bm, (short)0, c,
                                              false, false);
  }

  // epilogue: bias, per-column partial stats, tile store
  const float bv = bias[n0 + rc];
  float ls = 0.0f, lq = 0.0f;
#pragma unroll
  for (int r = 0; r < 8; ++r) {
    float v = c[r] + bv;
    c[r] = v;
    ls += v;
    lq += v * v;
  }
  const int rbase = (lane >> 4) * 8;        // C/D: VGPR r holds row r (+8 hi lanes)
#pragma unroll
  for (int r = 0; r < 8; ++r) {
    h[(size_t)(m0 + rbase + r) * COUT + n0 + rc] = c[r];
  }

  // lanes L and L+16 hold the same column -> fold, then LDS reduce, one atomic set
  ls += __shfl_xor(ls, 16, 32);
  lq += __shfl_xor(lq, 16, 32);

  __shared__ float rs[16], rq[16];
  if (threadIdx.x < 16) { rs[threadIdx.x] = 0.0f; rq[threadIdx.x] = 0.0f; }
  __syncthreads();
  if (lane < 16) {
    atomicAdd(&rs[rc], ls);
    atomicAdd(&rq[rc], lq);
  }
  __syncthreads();
  if (threadIdx.x < 16) {
    atomicAdd(&gsum[n0 + threadIdx.x], rs[threadIdx.x]);
    atomicAdd(&gsq [n0 + threadIdx.x], rq[threadIdx.x]);
  }
}

// ---------------------------------------------------------------------------
// Kernel 3: finalize BN -> per-channel affine coefficients a*h + b
// ---------------------------------------------------------------------------
__global__ void finalize_kernel(const float* __restrict__ gsum,
                                const float* __restrict__ gsq,
                                const float* __restrict__ gamma,
                                const float* __restrict__ beta,
                                float* __restrict__ cA, float* __restrict__ cB) {
  int c = threadIdx.x;
  const float n = (float)(NBATCH * NPTS);
  float mean = gsum[c] / n;
  float var  = gsq[c] / n - mean * mean;
  float a = gamma[c] * rsqrtf(var + EPSV);
  cA[c] = a;
  cB[c] = beta[c] - mean * a;
}

// ---------------------------------------------------------------------------
// Kernel 4: exact 16-NN per sampled point. Batch's points staged in LDS,
// one wave per query, 16 passes of lexicographic (d, idx) min via shuffles
// (matches lax.top_k tie ordering).
// ---------------------------------------------------------------------------
__global__ __launch_bounds__(256) void knn_kernel(const float* __restrict__ pos,
                                                  const int* __restrict__ fps_idx,
                                                  int* __restrict__ knn_idx) {
  __shared__ float sx[NPTS], sy[NPTS], sz[NPTS];
  const int b = blockIdx.y;
  const int t = threadIdx.x;
  const float* pb = pos + (size_t)b * NPTS * 3;
  for (int i = t; i < NPTS; i += 256) {
    sx[i] = pb[i * 3 + 0];
    sy[i] = pb[i * 3 + 1];
    sz[i] = pb[i * 3 + 2];
  }
  __syncthreads();

  const int lane = t & 31;
  const int wv = t >> 5;
  const int s = blockIdx.x * 8 + wv;
  const int q = fps_idx[b * NSAMP + s];
  const float qx = sx[q], qy = sy[q], qz = sz[q];

  int* out = knn_idx + ((size_t)b * NSAMP + s) * KNNK;
  float last_d = -1.0f;
  int   last_i = -1;
  for (int k = 0; k < KNNK; ++k) {
    float bd = 3.4e38f;
    int   bi = 0x7fffffff;
    for (int j = 0; j < NPTS / 32; ++j) {
      int i = j * 32 + lane;
      float dx = sx[i] - qx, dy = sy[i] - qy, dz = sz[i] - qz;
      float d = dx * dx + dy * dy + dz * dz;
      bool cand = (d > last_d) || (d == last_d && i > last_i);
      if (cand && (d < bd || (d == bd && i < bi))) { bd = d; bi = i; }
    }
#pragma unroll
    for (int off = 16; off > 0; off >>= 1) {
      float od = __shfl_xor(bd, off, 32);
      int   oi = __shfl_xor(bi, off, 32);
      if (od < bd || (od == bd && oi < bi)) { bd = od; bi = oi; }
    }
    if (lane == 0) out[k] = bi;
    last_d = bd;
    last_i = bi;
  }
}

// ---------------------------------------------------------------------------
// Kernel 5: gather 16 neighbor rows of h, apply folded BN affine per element,
// max over neighbors, ReLU once at the end (ReLU commutes with max).
// ---------------------------------------------------------------------------
__global__ __launch_bounds__(128) void gather_max_kernel(const float* __restrict__ h,
                                                         const int* __restrict__ knn_idx,
                                                         const float* __restrict__ cA,
                                                         const float* __restrict__ cB,
                                                         float* __restrict__ out) {
  const int bs = blockIdx.x;      // 0..8191
  const int b = bs >> 10;
  const int c = threadIdx.x;      // channel
  const float a = cA[c], bb = cB[c];
  const int* ki = knn_idx + (size_t)bs * KNNK;
  const float* hb = h + (size_t)b * NPTS * COUT;
  float m = -3.4e38f;
#pragma unroll
  for (int k = 0; k < KNNK; ++k) {
    int i = ki[k];
    float v = hb[(size_t)i * COUT + c];
    m = fmaxf(m, fmaf(v, a, bb));
  }
  out[(size_t)bs * COUT + c] = fmaxf(m, 0.0f);
}

// ---------------------------------------------------------------------------
// Kernel 6: gather sampled positions
// ---------------------------------------------------------------------------
__global__ void pos_gather_kernel(const float* __restrict__ pos,
                                  const int* __restrict__ fps_idx,
                                  float* __restrict__ out2) {
  int i = blockIdx.x * blockDim.x + threadIdx.x;  // 0..8191
  if (i >= NBATCH * NSAMP) return;
  int b = i >> 10;
  int idx = fps_idx[i];
  const float* p = pos + ((size_t)b * NPTS + idx) * 3;
  float* o = out2 + (size_t)i * 3;
  o[0] = p[0];
  o[1] = p[1];
  o[2] = p[2];
}

// ---------------------------------------------------------------------------
extern "C" void kernel_launch(void* const* d_in, const int* in_sizes, int n_in,
                              void* d_out, int out_size, void* d_ws, size_t ws_size,
                              hipStream_t stream) {
  const float* x     = (const float*)d_in[0];  // [8,4096,64]
  const float* pos   = (const float*)d_in[1];  // [8,4096,3]
  const float* W     = (const float*)d_in[2];  // [128,64]
  const float* bias  = (const float*)d_in[3];  // [128]
  const float* gamma = (const float*)d_in[4];  // [128]
  const float* beta  = (const float*)d_in[5];  // [128]
  // d_in[6] = n_sampling scalar (1024, fixed at compile time here)

  char* ws = (char*)d_ws;
  float* h      = (float*)(ws);                                   // 16 MB
  int*   fpsidx = (int*)(ws + (size_t)16777216);                  // 32 KB
  int*   knnidx = (int*)(ws + (size_t)16777216 + 32768);          // 512 KB
  float* gsum   = (float*)(ws + (size_t)17334272);                // 128 f32
  float* gsq    = gsum + COUT;
  float* cA     = gsum + 2 * COUT;
  float* cB     = gsum + 3 * COUT;

  float* out1 = (float*)d_out;                                    // [8,1024,128]
  float* out2 = out1 + (size_t)NBATCH * NSAMP * COUT;             // [8,1024,3]

  init_stats_kernel<<<1, COUT, 0, stream>>>(gsum, gsq);
  fps_kernel<<<NBATCH, 256, 0, stream>>>(pos, fpsidx);
  gemm_stats_kernel<<<dim3(COUT / 16, (NBATCH * NPTS) / 16 / 8), 256, 0, stream>>>(
      x, W, bias, h, gsum, gsq);
  finalize_kernel<<<1, COUT, 0, stream>>>(gsum, gsq, gamma, beta, cA, cB);
  knn_kernel<<<dim3(NSAMP / 8, NBATCH), 256, 0, stream>>>(pos, fpsidx, knnidx);
  gather_max_kernel<<<NBATCH * NSAMP, COUT, 0, stream>>>(h, knnidx, cA, cB, out1);
  pos_gather_kernel<<<(NBATCH * NSAMP + 255) / 256, 256, 0, stream>>>(pos, fpsidx, out2);

  (void)in_sizes; (void)n_in; (void)out_size; (void)ws_size;
}